// Aggregator_2121713844996
// MI455X (gfx1250) — compile-verified
//
#include <hip/hip_runtime.h>

#define N_NODES 200000
#define N_EDGES 12800000
#define DIM     128

typedef __attribute__((ext_vector_type(2))) float v2f;
typedef __attribute__((ext_vector_type(8))) float v8f;

// ---------------------------------------------------------------------------
// Hardware f32 atomic add (CDNA5 GLOBAL_ATOMIC_ADD_F32, no return -> STOREcnt).
// Inline asm guarantees we never fall back to a CAS loop: 1.6G atomics make
// this the single most important instruction in the workload.
// ---------------------------------------------------------------------------
__device__ __forceinline__ void hw_atomic_add_f32(float* addr, float val) {
    unsigned long long a = (unsigned long long)addr;
    asm volatile("global_atomic_add_f32 %0, %1, off"
                 :
                 : "v"(a), "v"(val)
                 : "memory");
}

// ---------------------------------------------------------------------------
// Kernel 1: zero the side/scatter accumulator (d_ws), float4-vectorized.
// ---------------------------------------------------------------------------
__global__ __launch_bounds__(256) void zero_kernel(float4* __restrict__ p, int n4) {
    int i = blockIdx.x * blockDim.x + threadIdx.x;
    if (i < n4) {
        float4 z; z.x = 0.f; z.y = 0.f; z.z = 0.f; z.w = 0.f;
        p[i] = z;
    }
}

// ---------------------------------------------------------------------------
// Kernel 2: SpMM scatter-add.  One wave32 per edge; lane L handles columns
// [4L, 4L+4).  Gather of ego[col] row is a fully-coalesced 512B float4 read
// (ego is 102 MB -> L2-resident on the 192 MB L2); scatter is 4 hardware
// global_atomic_add_f32 per lane.
// ---------------------------------------------------------------------------
__global__ __launch_bounds__(256) void spmm_kernel(
    const int*   __restrict__ erow,
    const int*   __restrict__ ecol,
    const float* __restrict__ eval_,
    const float* __restrict__ ego,
    float*       __restrict__ side) {
    long long gtid = (long long)blockIdx.x * blockDim.x + threadIdx.x;
    int e = (int)(gtid >> 5);
    if (e >= N_EDGES) return;
    int lane = (int)(gtid & 31);

    int   r = erow[e];
    int   c = ecol[e];
    float v = eval_[e];

    float4 m = *((const float4*)(ego + (long long)c * DIM) + lane);
    float* dst = side + (long long)r * DIM + lane * 4;
    hw_atomic_add_f32(dst + 0, v * m.x);
    hw_atomic_add_f32(dst + 1, v * m.y);
    hw_atomic_add_f32(dst + 2, v * m.z);
    hw_atomic_add_f32(dst + 3, v * m.w);
}

// ---------------------------------------------------------------------------
// Kernel 3: fused bi-interaction + two GEMMs via V_WMMA_F32_16X16X4_F32.
// Block = 256 threads = 8 waves; block handles 16 node rows.
// Wave w computes output columns [16w, 16w+16) for both GEMMs.
//
// A-matrix 16x4 f32 fragment layout (ISA 7.12.2):
//   lanes 0-15 : M = lane,     v2f = {A[M][k+0], A[M][k+1]}
//   lanes 16-31: M = lane-16,  v2f = {A[M][k+2], A[M][k+3]}
// B-matrix 4x16 fragment (row striped across lanes):
//   lanes 0-15 : N = lane,     v2f = {B[k+0][N], B[k+1][N]}
//   lanes 16-31: N = lane-16,  v2f = {B[k+2][N], B[k+3][N]}
// C/D 16x16 f32: VGPR v holds M = v (lanes 0-15) / M = v+8 (lanes 16-31).
// ---------------------------------------------------------------------------
__global__ __launch_bounds__(256) void dense_kernel(
    const float* __restrict__ ego,
    const float* __restrict__ side,
    const float* __restrict__ W1,
    const float* __restrict__ b1,
    const float* __restrict__ W2,
    const float* __restrict__ b2,
    float*       __restrict__ out) {
    // stride 132 -> bank = (row*132 + k) % 64 = (4*row + k) % 64: conflict-free
    __shared__ float s_add [16][132];
    __shared__ float s_prod[16][132];

    int tid = threadIdx.x;
    long long r0 = (long long)blockIdx.x * 16;

    // Stage add = ego + side, prod = ego * side (coalesced, 8 elems/thread).
    for (int i = tid; i < 16 * DIM; i += 256) {
        int row = i >> 7;
        int col = i & 127;
        float e = ego [(r0 + row) * DIM + col];
        float s = side[(r0 + row) * DIM + col];
        s_add [row][col] = e + s;
        s_prod[row][col] = e * s;
    }
    __syncthreads();

    int wave  = tid >> 5;
    int lane  = tid & 31;
    int col   = wave * 16 + (lane & 15);   // output column this lane owns
    int mrow  = lane & 15;                 // A-fragment row this lane owns
    int khalf = (lane < 16) ? 0 : 2;       // K sub-offset for this half-wave

    v8f acc1 = {};  // add_emb @ W1 tile
    v8f acc2 = {};  // prod_emb @ W2 tile

    for (int k = 0; k < DIM; k += 4) {
        int kk = k + khalf;
        v2f a1, a2, wb1, wb2;
        a1.x  = s_add [mrow][kk];     a1.y  = s_add [mrow][kk + 1];
        a2.x  = s_prod[mrow][kk];     a2.y  = s_prod[mrow][kk + 1];
        wb1.x = W1[kk * DIM + col];   wb1.y = W1[(kk + 1) * DIM + col];
        wb2.x = W2[kk * DIM + col];   wb2.y = W2[(kk + 1) * DIM + col];
        acc1 = __builtin_amdgcn_wmma_f32_16x16x4_f32(
            false, a1, false, wb1, (short)0, acc1, false, false);
        acc2 = __builtin_amdgcn_wmma_f32_16x16x4_f32(
            false, a2, false, wb2, (short)0, acc2, false, false);
    }

    float bb1 = b1[col];
    float bb2 = b2[col];
    int mbase = (lane < 16) ? 0 : 8;

    #pragma unroll
    for (int v = 0; v < 8; ++v) {
        float x1 = acc1[v] + bb1;  x1 = (x1 > 0.f) ? x1 : 0.01f * x1;
        float x2 = acc2[v] + bb2;  x2 = (x2 > 0.f) ? x2 : 0.01f * x2;
        out[(r0 + mbase + v) * DIM + col] = x1 + x2;
    }
}

// ---------------------------------------------------------------------------
extern "C" void kernel_launch(void* const* d_in, const int* in_sizes, int n_in,
                              void* d_out, int out_size, void* d_ws, size_t ws_size,
                              hipStream_t stream) {
    const int*   erow  = (const int*)  d_in[0];
    const int*   ecol  = (const int*)  d_in[1];
    const float* eval_ = (const float*)d_in[2];
    const float* ego   = (const float*)d_in[3];
    const float* W1    = (const float*)d_in[4];
    const float* b1    = (const float*)d_in[5];
    const float* W2    = (const float*)d_in[6];
    const float* b2    = (const float*)d_in[7];
    float*       out   = (float*)d_out;
    float*       side  = (float*)d_ws;   // N_NODES * DIM floats = 102.4 MB

    // 1) zero accumulator
    int n4 = N_NODES * DIM / 4;
    zero_kernel<<<(n4 + 255) / 256, 256, 0, stream>>>((float4*)side, n4);

    // 2) SpMM scatter-add: one wave per edge
    long long total_threads = (long long)N_EDGES * 32;
    int blocks = (int)((total_threads + 255) / 256);
    spmm_kernel<<<blocks, 256, 0, stream>>>(erow, ecol, eval_, ego, side);

    // 3) fused bi-interaction + GEMMs (200000 / 16 = 12500 blocks)
    dense_kernel<<<N_NODES / 16, 256, 0, stream>>>(ego, side, W1, b1, W2, b2, out);
}